// Group_41824391528658
// MI455X (gfx1250) — compile-verified
//
#include <hip/hip_runtime.h>
#include <hip/hip_bf16.h>
#include <stdint.h>

// Problem constants (match reference)
#define BATCH   16
#define NPTS    16384
#define NGROUP  512
#define KNN     32
#define SEL_STRIDE (NPTS / NGROUP)   // 32

// Tiling
#define GPT     16        // groups per workgroup == WMMA M
#define TPB     256       // 8 wave32 per workgroup
#define NWAVE   (TPB / 32)
#define NTILE   (NPTS / 16)          // 16-point WMMA tiles along N

// Selection: log2-scale histogram on float bits (exp + top-2 mantissa bits).
// bits(d) >> 21 for finite d >= 0 is in [0, 1020] -> 1024 bins, no clamp.
#define HBINS   1024
#define HSHIFT  21
#define CAP     256       // per-group candidate capacity (expected ~42)

typedef float v2f __attribute__((ext_vector_type(2)));
typedef float v8f __attribute__((ext_vector_type(8)));

// ---------------------------------------------------------------------------
// Shared helpers
// ---------------------------------------------------------------------------

// Load 16 group centers into LDS as (cx,cy,cz,|c|^2). Call with tid<GPT, then sync.
__device__ __forceinline__ void load_centers(const float* __restrict__ xyzb,
                                             int g0, int tid, float* cen) {
    if (tid < GPT) {
        const float* cp = xyzb + (size_t)(g0 + tid) * SEL_STRIDE * 3;
        float cx = cp[0], cy = cp[1], cz = cp[2];
        cen[tid * 4 + 0] = cx;
        cen[tid * 4 + 1] = cy;
        cen[tid * 4 + 2] = cz;
        cen[tid * 4 + 3] = __builtin_fmaf(cx, cx, __builtin_fmaf(cy, cy, cz * cz));
    }
}

// Build this lane's A fragment for V_WMMA_F32_16X16X4_F32:
//   A row m = (-2cx, -2cy, -2cz, 1.0)
// Layout (32-bit A 16x4): lanes 0-15 hold K=0,1 ; lanes 16-31 hold K=2,3.
__device__ __forceinline__ v2f build_afrag(const float* cen, int mi, int half) {
    v2f a;
    if (half == 0) {
        a.x = -2.0f * cen[mi * 4 + 0];
        a.y = -2.0f * cen[mi * 4 + 1];
    } else {
        a.x = -2.0f * cen[mi * 4 + 2];
        a.y = 1.0f;
    }
    return a;
}

// C fragment = |c|^2 broadcast per row: lane l, VGPR r holds row m = r + 8*(l>=16).
__device__ __forceinline__ v8f build_cfrag(const float* cen, int half) {
    v8f c;
#pragma unroll
    for (int r = 0; r < 8; ++r) c[r] = cen[(r + 8 * half) * 4 + 3];
    return c;
}

// Monotone distance key with negative clamp, in integer domain:
// for finite d, int bits < 0 <=> d < 0, and non-negative floats have bits < 2^31,
// so a single signed max replaces fmaxf's canonicalize+max pair.
__device__ __forceinline__ unsigned dist_bits(float d) {
    const int b = __float_as_int(d);
    return (unsigned)(b < 0 ? 0 : b);   // v_max_i32 b, 0
}

// Build this lane's B fragment for one 16-point tile: col n = (x, y, z, |p|^2).
__device__ __forceinline__ v2f load_bfrag(const float* __restrict__ pp, int half) {
    const float x = pp[0], y = pp[1], z = pp[2];
    const float s = __builtin_fmaf(x, x, __builtin_fmaf(y, y, z * z));
    v2f b;
    b.x = half ? z : x;
    b.y = half ? s : y;
    return b;
}

// ---------------------------------------------------------------------------
// Kernel 1: per-group log2-distance histograms -> threshold bin for top-32
// grid = BATCH * (NGROUP/GPT) blocks of TPB threads.
// Dynamic LDS: hist[GPT*HBINS] u32 (64KB) + cen[GPT*4] f32
// ---------------------------------------------------------------------------
__global__ __launch_bounds__(TPB, 1)
void knn_hist_kernel(const float* __restrict__ xyz, unsigned* __restrict__ thr) {
    extern __shared__ unsigned char smem[];
    unsigned* hist = (unsigned*)smem;                                 // GPT*HBINS
    float*    cen  = (float*)(smem + GPT * HBINS * sizeof(unsigned)); // GPT*4

    const int bb = blockIdx.x / (NGROUP / GPT);
    const int g0 = (blockIdx.x % (NGROUP / GPT)) * GPT;
    const float* xyzb = xyz + (size_t)bb * NPTS * 3;

    const int tid  = threadIdx.x;
    const int lane = tid & 31;
    const int wave = tid >> 5;
    const int half = lane >> 4;
    const int mi   = lane & 15;

    for (int i = tid; i < GPT * HBINS; i += TPB) hist[i] = 0u;
    load_centers(xyzb, g0, tid, cen);
    __syncthreads();

    const v2f afrag = build_afrag(cen, mi, half);
    const v8f cfrag = build_cfrag(cen, half);   // |c|^2 row bias via WMMA C operand

    // Stream all N points in 16-wide tiles; one WMMA per tile per wave.
    // D = (-2c . p) + |p|^2 + |c|^2  == full squared distance, directly.
    for (int t = wave; t < NTILE; t += NWAVE) {
        const float* pp = xyzb + (size_t)(t * 16 + mi) * 3;
        __builtin_prefetch(pp + (size_t)NWAVE * 16 * 3, 0, 1);  // next iter's tile
        const v2f bfrag = load_bfrag(pp, half);

        const v8f acc = __builtin_amdgcn_wmma_f32_16x16x4_f32(
            /*neg_a=*/false, afrag, /*neg_b=*/false, bfrag,
            /*c_mod=*/(short)0, cfrag, /*reuse_a=*/false, /*reuse_b=*/false);

#pragma unroll
        for (int r = 0; r < 8; ++r) {
            const unsigned bin = dist_bits(acc[r]) >> HSHIFT;
            atomicAdd(&hist[(r + 8 * half) * HBINS + bin], 1u);
        }
    }
    __syncthreads();

    // One thread per group: smallest bin b with cum(<=b) >= 32
    if (tid < GPT) {
        unsigned cum = 0;
        int tb = HBINS - 1;
        for (int i = 0; i < HBINS; ++i) {
            cum += hist[tid * HBINS + i];
            if (cum >= KNN) { tb = i; break; }
        }
        thr[(size_t)bb * NGROUP + g0 + tid] = (unsigned)tb;
    }
}

// ---------------------------------------------------------------------------
// Kernel 2: recompute distances (same WMMA tiles), gather candidates under
// threshold into LDS, rank-sort, and write neighborhood/center/ids.
// Dynamic LDS: keys[GPT*CAP] u64 + cnt[GPT] u32 + cen[GPT*4] f32 + thrs[GPT] u32
// ---------------------------------------------------------------------------
__global__ __launch_bounds__(TPB, 1)
void knn_select_kernel(const float* __restrict__ xyz,
                       const unsigned* __restrict__ thr,
                       float* __restrict__ out_nb,
                       float* __restrict__ out_c,
                       int* __restrict__ out_ids) {
    extern __shared__ unsigned char smem[];
    unsigned long long* keys = (unsigned long long*)smem;            // GPT*CAP
    unsigned* cnt  = (unsigned*)(smem + GPT * CAP * 8);              // GPT
    float*    cen  = (float*)((unsigned char*)cnt + GPT * 4);        // GPT*4
    unsigned* thrs = (unsigned*)((unsigned char*)cen + GPT * 16);    // GPT

    const int bb = blockIdx.x / (NGROUP / GPT);
    const int g0 = (blockIdx.x % (NGROUP / GPT)) * GPT;
    const float* xyzb = xyz + (size_t)bb * NPTS * 3;

    const int tid  = threadIdx.x;
    const int lane = tid & 31;
    const int wave = tid >> 5;
    const int half = lane >> 4;
    const int mi   = lane & 15;

    if (tid < GPT) {
        cnt[tid]  = 0u;
        thrs[tid] = thr[(size_t)bb * NGROUP + g0 + tid];
    }
    load_centers(xyzb, g0, tid, cen);
    __syncthreads();

    const v2f afrag = build_afrag(cen, mi, half);
    const v8f cfrag = build_cfrag(cen, half);
    unsigned thrbits[8];   // candidate iff bits(d) < thrbits  <=>  bin(d) <= thr
#pragma unroll
    for (int r = 0; r < 8; ++r) thrbits[r] = (thrs[r + 8 * half] + 1u) << HSHIFT;

    for (int t = wave; t < NTILE; t += NWAVE) {
        const int p0 = t * 16 + mi;
        const float* pp = xyzb + (size_t)p0 * 3;
        __builtin_prefetch(pp + (size_t)NWAVE * 16 * 3, 0, 1);
        const v2f bfrag = load_bfrag(pp, half);

        const v8f acc = __builtin_amdgcn_wmma_f32_16x16x4_f32(
            false, afrag, false, bfrag, (short)0, cfrag, false, false);

#pragma unroll
        for (int r = 0; r < 8; ++r) {
            const unsigned bits = dist_bits(acc[r]);
            if (bits < thrbits[r]) {
                const int m = r + 8 * half;
                const unsigned slot = atomicAdd(&cnt[m], 1u);
                if (slot < CAP) {
                    keys[m * CAP + slot] =
                        ((unsigned long long)bits << 32) |
                        (unsigned long long)(unsigned)p0;
                }
            }
        }
    }
    __syncthreads();

    // Rank sort & write: each wave handles 2 groups.
#pragma unroll
    for (int mIter = 0; mIter < 2; ++mIter) {
        const int m = wave + mIter * NWAVE;      // 0..15
        const int g = g0 + m;
        unsigned mcount = cnt[m];
        if (mcount > CAP) mcount = CAP;
        const unsigned long long* kk = &keys[m * CAP];
        const float cx = cen[m * 4 + 0];
        const float cy = cen[m * 4 + 1];
        const float cz = cen[m * 4 + 2];

        for (unsigned j = lane; j < mcount; j += 32) {
            const unsigned long long kj = kk[j];
            int rank = 0;
            for (unsigned q = 0; q < mcount; ++q) rank += (kk[q] < kj) ? 1 : 0;
            if (rank < KNN) {
                const unsigned idx = (unsigned)(kj & 0xFFFFFFFFull);
                const float* pp = xyzb + (size_t)idx * 3;
                float* o = out_nb + (((size_t)bb * NGROUP + g) * KNN + rank) * 3;
                o[0] = pp[0] - cx;
                o[1] = pp[1] - cy;
                o[2] = pp[2] - cz;
            }
        }
        // Safety: if (pathologically) fewer than 32 candidates, zero-fill.
        for (unsigned rf = mcount + lane; rf < KNN; rf += 32) {
            float* o = out_nb + (((size_t)bb * NGROUP + g) * KNN + rf) * 3;
            o[0] = 0.0f; o[1] = 0.0f; o[2] = 0.0f;
        }
        if (lane == 0) {
            float* oc = out_c + ((size_t)bb * NGROUP + g) * 3;
            oc[0] = cx; oc[1] = cy; oc[2] = cz;
            out_ids[(size_t)bb * NGROUP + g] = g * SEL_STRIDE;
        }
    }
}

// ---------------------------------------------------------------------------
// Launch
// ---------------------------------------------------------------------------
extern "C" void kernel_launch(void* const* d_in, const int* in_sizes, int n_in,
                              void* d_out, int out_size, void* d_ws, size_t ws_size,
                              hipStream_t stream) {
    (void)in_sizes; (void)n_in; (void)out_size; (void)ws_size;
    const float* xyz = (const float*)d_in[0];

    float* out     = (float*)d_out;
    float* out_nb  = out;                                            // 786432 f32
    float* out_c   = out + (size_t)BATCH * NGROUP * KNN * 3;         // 24576 f32
    int*   out_ids = (int*)(out + (size_t)BATCH * NGROUP * (KNN * 3 + 3)); // 8192 i32

    unsigned* thr = (unsigned*)d_ws;   // BATCH*NGROUP u32 = 32KB

    const dim3 grid(BATCH * (NGROUP / GPT));   // 512 workgroups
    const dim3 block(TPB);

    const size_t smem1 = (size_t)GPT * HBINS * sizeof(unsigned) + GPT * 4 * sizeof(float);
    const size_t smem2 = (size_t)GPT * CAP * 8 + GPT * 4 + GPT * 16 + GPT * 4;

    knn_hist_kernel<<<grid, block, smem1, stream>>>(xyz, thr);
    knn_select_kernel<<<grid, block, smem2, stream>>>(xyz, thr, out_nb, out_c, out_ids);
}